// GCNBlock_66812511257309
// MI455X (gfx1250) — compile-verified
//
#include <hip/hip_runtime.h>

#define DFEAT 128

typedef __attribute__((ext_vector_type(2))) float v2f;
typedef __attribute__((ext_vector_type(8))) float v8f;

// ---------------------------------------------------------------------------
// Degree: deg[i] = 1 (self loop) + #incoming edges
// ---------------------------------------------------------------------------
__global__ void init_deg_kernel(float* __restrict__ deg, int n) {
    int i = blockIdx.x * blockDim.x + threadIdx.x;
    if (i < n) deg[i] = 1.0f;
}

__global__ void count_deg_kernel(const long long* __restrict__ dst,
                                 float* __restrict__ deg, int e) {
    int i = blockIdx.x * blockDim.x + threadIdx.x;
    if (i < e) atomicAdd(&deg[(int)dst[i]], 1.0f);
}

__global__ void rsqrt_kernel(const float* __restrict__ deg,
                             float* __restrict__ dinv, int n) {
    int i = blockIdx.x * blockDim.x + threadIdx.x;
    if (i < n) dinv[i] = rsqrtf(deg[i]);
}

// ---------------------------------------------------------------------------
// h = x @ W via V_WMMA_F32_16X16X4_F32.
// Block: 256 threads = 8 waves. Each block handles 16-row strips (grid-stride).
// LDS: full W (128x128 f32, 64KB) + 16x128 A panel padded to stride 129.
// Wave w computes the 16x16 tile at columns [16w, 16w+16).
// A layout (16x4 f32):  lanes 0-15 -> {K=k,k+1}, lanes 16-31 -> {K=k+2,k+3}
// B layout (4x16 f32):  lanes 0-15 -> rows k,k+1 at col N=lane,
//                       lanes 16-31 -> rows k+2,k+3 at col N=lane-16
// C/D layout: VGPR v -> row v (lanes 0-15) / row v+8 (lanes 16-31), N=lane%16
// ---------------------------------------------------------------------------
__global__ void gemm_wmma_kernel(const float* __restrict__ x,
                                 const float* __restrict__ W,
                                 float* __restrict__ h, int n) {
    __shared__ float sW[DFEAT * DFEAT];   // 64 KB
    __shared__ float sA[16 * 129];        // padded: bank-conflict-free col reads

    const int tid    = threadIdx.x;
    const int wave   = tid >> 5;
    const int lane   = tid & 31;
    const int lane16 = lane & 15;
    const bool hi    = lane >= 16;
    const int c0     = wave * 16;

    // Stage W once per block (coalesced).
    for (int i = tid; i < DFEAT * DFEAT; i += 256) sW[i] = W[i];

    const int nstrips = (n + 15) >> 4;
    for (int strip = blockIdx.x; strip < nstrips; strip += gridDim.x) {
        const int r0   = strip << 4;
        const int rows = min(16, n - r0);

        __syncthreads();  // protect sA from previous iteration (also orders sW)
        for (int i = tid; i < 16 * DFEAT; i += 256) {
            int r = i >> 7, c = i & (DFEAT - 1);
            sA[r * 129 + c] = (r < rows) ? x[(size_t)(r0 + r) * DFEAT + c] : 0.0f;
        }
        __syncthreads();

        v8f acc = {};
        #pragma unroll
        for (int k = 0; k < DFEAT; k += 4) {
            const int ka = hi ? k + 2 : k;
            v2f a, b;
            a.x = sA[lane16 * 129 + ka];
            a.y = sA[lane16 * 129 + ka + 1];
            b.x = sW[ka * DFEAT + c0 + lane16];
            b.y = sW[(ka + 1) * DFEAT + c0 + lane16];
            acc = __builtin_amdgcn_wmma_f32_16x16x4_f32(
                false, a, false, b, (short)0, acc, false, false);
        }

        #pragma unroll
        for (int v = 0; v < 8; ++v) {
            const int r = hi ? v + 8 : v;
            if (r < rows)
                h[(size_t)(r0 + r) * DFEAT + c0 + lane16] = acc[v];
        }
    }
}

// ---------------------------------------------------------------------------
// agg[i,:] = h[i,:] * dinv[i]^2   (self-loop contribution), agg lives in d_out
// ---------------------------------------------------------------------------
__global__ void init_agg_kernel(const float* __restrict__ h,
                                const float* __restrict__ dinv,
                                float* __restrict__ agg, int n) {
    long long idx = (long long)blockIdx.x * blockDim.x + threadIdx.x;  // float4 idx
    long long total = (long long)n * (DFEAT / 4);
    if (idx < total) {
        int node = (int)(idx >> 5);
        float s = dinv[node];
        s *= s;
        float4 v = ((const float4*)h)[idx];
        v.x *= s; v.y *= s; v.z *= s; v.w *= s;
        ((float4*)agg)[idx] = v;
    }
}

// ---------------------------------------------------------------------------
// One wave per edge: lane owns a float4 slice of the 128-wide feature.
// coef = dinv[src]*dinv[dst]; agg[dst] += h[src]*coef via f32 atomics (L2).
// ---------------------------------------------------------------------------
__global__ void scatter_kernel(const float* __restrict__ h,
                               const float* __restrict__ dinv,
                               const long long* __restrict__ src,
                               const long long* __restrict__ dst,
                               float* __restrict__ agg, int e) {
    const int lane  = threadIdx.x & 31;
    long long wave  = ((long long)blockIdx.x * blockDim.x + threadIdx.x) >> 5;
    long long nwave = ((long long)gridDim.x * blockDim.x) >> 5;
    for (long long ed = wave; ed < e; ed += nwave) {
        const int s = (int)src[ed];
        const int d = (int)dst[ed];
        const float coef = dinv[s] * dinv[d];
        float4 m = ((const float4*)(h + (size_t)s * DFEAT))[lane];
        float* ad = agg + (size_t)d * DFEAT + lane * 4;
        atomicAdd(ad + 0, m.x * coef);
        atomicAdd(ad + 1, m.y * coef);
        atomicAdd(ad + 2, m.z * coef);
        atomicAdd(ad + 3, m.w * coef);
    }
}

// ---------------------------------------------------------------------------
// out = relu(agg + b) + x, in place on d_out
// ---------------------------------------------------------------------------
__global__ void finalize_kernel(const float* __restrict__ x,
                                const float* __restrict__ b,
                                float* __restrict__ agg_out, int n) {
    long long idx = (long long)blockIdx.x * blockDim.x + threadIdx.x;  // float4 idx
    long long total = (long long)n * (DFEAT / 4);
    if (idx < total) {
        float4 a  = ((float4*)agg_out)[idx];
        float4 bb = ((const float4*)b)[idx & 31];
        float4 xx = ((const float4*)x)[idx];
        a.x = fmaxf(a.x + bb.x, 0.0f) + xx.x;
        a.y = fmaxf(a.y + bb.y, 0.0f) + xx.y;
        a.z = fmaxf(a.z + bb.z, 0.0f) + xx.z;
        a.w = fmaxf(a.w + bb.w, 0.0f) + xx.w;
        ((float4*)agg_out)[idx] = a;
    }
}

extern "C" void kernel_launch(void* const* d_in, const int* in_sizes, int n_in,
                              void* d_out, int out_size, void* d_ws, size_t ws_size,
                              hipStream_t stream) {
    const float*     x  = (const float*)d_in[0];
    const float*     W  = (const float*)d_in[1];
    const float*     b  = (const float*)d_in[2];
    const long long* ei = (const long long*)d_in[3];

    const int n = in_sizes[0] / DFEAT;   // 100000
    const int e = in_sizes[3] / 2;       // 1600000
    const long long* src = ei;
    const long long* dst = ei + e;

    // Workspace layout: deg[n] | dinv[n] | h[n*128]
    float* deg  = (float*)d_ws;
    float* dinv = deg + n;
    float* h    = dinv + n;              // byte offset 800000, 16B-aligned
    float* agg  = (float*)d_out;         // accumulate directly into output

    const int T = 256;

    init_deg_kernel<<<(n + T - 1) / T, T, 0, stream>>>(deg, n);
    count_deg_kernel<<<(e + T - 1) / T, T, 0, stream>>>(dst, deg, e);
    rsqrt_kernel<<<(n + T - 1) / T, T, 0, stream>>>(deg, dinv, n);

    gemm_wmma_kernel<<<2048, T, 0, stream>>>(x, W, h, n);

    long long q = (long long)n * (DFEAT / 4);
    init_agg_kernel<<<(unsigned)((q + T - 1) / T), T, 0, stream>>>(h, dinv, agg, n);

    scatter_kernel<<<16384, T, 0, stream>>>(h, dinv, src, dst, agg, e);

    finalize_kernel<<<(unsigned)((q + T - 1) / T), T, 0, stream>>>(x, b, agg, n);

    // Tuple output: append edge_index (pass-through) after out[n*128].
    long long out_floats = (long long)n * DFEAT;
    long long ei_floats  = (long long)e * 4;  // 2*e int64 == 4*e float-words
    if ((long long)out_size >= out_floats + ei_floats) {
        hipMemcpyAsync((char*)d_out + out_floats * sizeof(float), ei,
                       (size_t)(2 * (long long)e) * sizeof(long long),
                       hipMemcpyDeviceToDevice, stream);
    }
}